// LightGCN_20109036880396
// MI455X (gfx1250) — compile-verified
//
#include <hip/hip_runtime.h>
#include <hip/hip_bf16.h>
#include <stdint.h>

// ---------------------------------------------------------------------------
// LightGCN bipartite propagation on MI455X (gfx1250).
//   USER=100000, ITEM=50000, DIM=64, NNZ=6400000, LAYER=3
// Memory-bound sparse gather/scatter (0.25 flop/byte) -> optimize data path:
//   * 38.4MB embedding level fits in 192MB L2 -> gathers/atomics are L2 hits
//   * 16 lanes x float4 per edge -> coalesced 256B row gather (global_load_b128)
//   * hardware f32 atomics for scatter-add (global_atomic_add_f32, L2 RMW)
//   * TDM tensor_load_to_lds streams COO edge tiles into LDS (TENSORcnt)
// ---------------------------------------------------------------------------

#define N_USER  100000
#define N_ITEM  50000
#define N_NODE  150000              // USER + ITEM
#define DIM     64
#define SZ      (N_NODE * DIM)      // 9,600,000 floats per level
#define NNZ     6400000
#define TILE    1024                // edges staged per block via TDM

typedef __attribute__((ext_vector_type(4))) unsigned int tdm_u32x4;
typedef __attribute__((ext_vector_type(8))) int          tdm_i32x8;
typedef __attribute__((ext_vector_type(4))) int          tdm_i32x4;

// Issue one TDM 1-D copy: n_elems dwords from gaddr -> LDS byte offset lds_off.
// D# layout per CDNA5 ISA ch.8 (group0 128b, group1 256b, groups2/3 zero).
// 6-arg builtin form (amdgpu-toolchain clang-23 / therock-10.0 headers).
static __device__ __forceinline__ void tdm_load_1d_dwords(unsigned lds_off,
                                                          const void* gaddr,
                                                          unsigned n_elems,
                                                          unsigned tensor_len) {
    unsigned long long ga = (unsigned long long)(uintptr_t)gaddr;
    tdm_u32x4 g0;
    g0.x = 1u;                                            // count=1, user desc
    g0.y = lds_off;                                       // lds_addr (bytes)
    g0.z = (unsigned)(ga & 0xffffffffull);                // global_addr[31:0]
    g0.w = (unsigned)((ga >> 32) & 0x01ffffffull)         // global_addr[56:32]
         | (2u << 30);                                    // type=2 ("image")
    tdm_i32x8 g1;
    g1[0] = (int)(2u << 16);                              // data_size=2 -> 4B
    g1[1] = (int)((tensor_len & 0xffffu) << 16);          // tensor_dim0[15:0]
    g1[2] = (int)(((tensor_len >> 16) & 0xffffu)          // tensor_dim0[31:16]
         | (1u << 16));                                   // tensor_dim1 = 1
    g1[3] = (int)((n_elems & 0xffffu) << 16);             // tile_dim0
    g1[4] = 0;                                            // tile_dim1/2 unused
    g1[5] = (int)tensor_len;                              // dim0_stride[31:0]
    g1[6] = 0;                                            // dim0_stride[47:32]
    g1[7] = 0;
    tdm_i32x4 gz4 = {0, 0, 0, 0};
    tdm_i32x8 gz8 = {0, 0, 0, 0, 0, 0, 0, 0};
    __builtin_amdgcn_tensor_load_to_lds(g0, g1, gz4, gz4, gz8, 0);
}

// -------------------------- kernel 1: init ---------------------------------
// lats[0] = gcn[0] = embeds ; gcn[1..3] = 0  (float4 vectorized over SZ/4)
__global__ void lgcn_init(const float* __restrict__ embeds,
                          float* __restrict__ out) {
    size_t i = (size_t)blockIdx.x * blockDim.x + threadIdx.x;  // float4 index
    if (i >= (size_t)SZ / 4) return;
    float4 v = ((const float4*)embeds)[i];
    ((float4*)(out + 0 * (size_t)SZ))[i] = v;                  // lats[0]
    ((float4*)(out + 4 * (size_t)SZ))[i] = v;                  // gcn[0]
    float4 z = make_float4(0.f, 0.f, 0.f, 0.f);
    ((float4*)(out + 5 * (size_t)SZ))[i] = z;                  // gcn[1]
    ((float4*)(out + 6 * (size_t)SZ))[i] = z;                  // gcn[2]
    ((float4*)(out + 7 * (size_t)SZ))[i] = z;                  // gcn[3]
}

// ------------------------ kernel 2: edge scatter ---------------------------
// For one layer: gcn_user[r] += v * lat_prev_item[c]; gcn_item[c] += v * lat_prev_user[r].
// 256 threads/block, 16 lanes per edge (float4 each), TILE edges per block
// staged in LDS by the Tensor Data Mover.
__global__ __launch_bounds__(256) void lgcn_scatter(
        const int*   __restrict__ rows,
        const int*   __restrict__ cols,
        const float* __restrict__ vals,
        const float* __restrict__ lat_prev,
        float*       __restrict__ gcn) {
    __shared__ __align__(16) unsigned char smem[TILE * 12];
    int*   srows = (int*)(smem);
    int*   scols = (int*)(smem + TILE * 4);
    float* svals = (float*)(smem + TILE * 8);

    const unsigned tileStart = (unsigned)blockIdx.x * TILE;
    const unsigned remaining = NNZ - tileStart;
    const unsigned n = remaining < TILE ? remaining : TILE;

    // Wave 0 issues three async TDM copies (EXEC-independent, per-wave ops),
    // waits TENSORcnt==0, then the workgroup barrier publishes LDS to all waves.
    if (threadIdx.x < 32) {
        tdm_load_1d_dwords(0u,        rows + tileStart, n, remaining);
        tdm_load_1d_dwords(TILE * 4u, cols + tileStart, n, remaining);
        tdm_load_1d_dwords(TILE * 8u, vals + tileStart, n, remaining);
        __builtin_amdgcn_s_wait_tensorcnt(0);
    }
    // The TDM builtin carries no pointer to smem, so tell the optimizer the
    // LDS buffer has been written (otherwise the ds_loads fold to undef).
    {
        unsigned char* p = smem;
        asm volatile("" : "+r"(p) : : "memory");
    }
    __syncthreads();

    const int lane = (int)(threadIdx.x & 15);   // which float4 of the 64-dim row
    const int eg   = (int)(threadIdx.x >> 4);   // edge sub-group 0..15

#pragma unroll 4
    for (int t = 0; t < TILE / 16; ++t) {
        const int el = t * 16 + eg;
        const unsigned e = tileStart + (unsigned)el;
        if (e >= NNZ) break;

        const int   r = srows[el];
        const int   c = scols[el];
        const float v = svals[el];

        // Prefetch next edge-group's gather rows (global_prefetch_b8).
        const int el2 = el + 16;
        if (el2 < (int)n) {
            __builtin_prefetch(lat_prev + (size_t)srows[el2] * DIM + lane * 4, 0, 0);
            __builtin_prefetch(lat_prev + ((size_t)N_USER + scols[el2]) * DIM + lane * 4, 0, 0);
        }

        const float4 uv = ((const float4*)(lat_prev + (size_t)r * DIM))[lane];
        const float4 iv = ((const float4*)(lat_prev + ((size_t)N_USER + c) * DIM))[lane];

        float* udst = gcn + (size_t)r * DIM + lane * 4;            // += v * item row
        float* idst = gcn + ((size_t)N_USER + c) * DIM + lane * 4; // += v * user row
        unsafeAtomicAdd(udst + 0, v * iv.x);
        unsafeAtomicAdd(udst + 1, v * iv.y);
        unsafeAtomicAdd(udst + 2, v * iv.z);
        unsafeAtomicAdd(udst + 3, v * iv.w);
        unsafeAtomicAdd(idst + 0, v * uv.x);
        unsafeAtomicAdd(idst + 1, v * uv.y);
        unsafeAtomicAdd(idst + 2, v * uv.z);
        unsafeAtomicAdd(idst + 3, v * uv.w);
    }
}

// ------------------------ kernel 3: residual add ---------------------------
// lats[l] = lats[l-1] + gcn[l]
__global__ void lgcn_resid(float* __restrict__ lat_out,
                           const float* __restrict__ lat_prev,
                           const float* __restrict__ gcn) {
    size_t i = (size_t)blockIdx.x * blockDim.x + threadIdx.x;  // float4 index
    if (i >= (size_t)SZ / 4) return;
    float4 a = ((const float4*)lat_prev)[i];
    float4 b = ((const float4*)gcn)[i];
    float4 r = make_float4(a.x + b.x, a.y + b.y, a.z + b.z, a.w + b.w);
    ((float4*)lat_out)[i] = r;
}

// ---------------------------------------------------------------------------
extern "C" void kernel_launch(void* const* d_in, const int* in_sizes, int n_in,
                              void* d_out, int out_size, void* d_ws, size_t ws_size,
                              hipStream_t stream) {
    (void)in_sizes; (void)n_in; (void)out_size; (void)d_ws; (void)ws_size;

    const int*   rows   = (const int*)d_in[0];
    const int*   cols   = (const int*)d_in[1];
    const float* vals   = (const float*)d_in[2];
    const float* embeds = (const float*)d_in[3];
    float*       out    = (float*)d_out;

    // d_out layout: lats[0..3] then gcn_lats[0..3], each SZ floats.
    float* lat[4];
    float* gcn[4];
    for (int l = 0; l < 4; ++l) {
        lat[l] = out + (size_t)l * SZ;
        gcn[l] = out + (size_t)(4 + l) * SZ;
    }

    const int vecBlocks = (SZ / 4 + 255) / 256;           // 9375
    const int edgeBlocks = (NNZ + TILE - 1) / TILE;       // 6250

    lgcn_init<<<vecBlocks, 256, 0, stream>>>(embeds, out);

    for (int l = 1; l <= 3; ++l) {
        lgcn_scatter<<<edgeBlocks, 256, 0, stream>>>(rows, cols, vals,
                                                     lat[l - 1], gcn[l]);
        lgcn_resid<<<vecBlocks, 256, 0, stream>>>(lat[l], lat[l - 1], gcn[l]);
    }
}